// kDivFree_21500606284258
// MI455X (gfx1250) — compile-verified
//
#include <hip/hip_runtime.h>
#include <math.h>

typedef float v2f __attribute__((ext_vector_type(2)));
typedef float v8f __attribute__((ext_vector_type(8)));

#define HID   100
#define NTILE 7   // ceil(100/16) hidden tiles (padded to 112, pads are zeroed)

__device__ __forceinline__ float fast_tanh(float x) {
#if __has_builtin(__builtin_amdgcn_tanhf)
  return __builtin_amdgcn_tanhf(x);          // v_tanh_f32 (gfx1250 TRANS unit)
#elif __has_builtin(__builtin_amdgcn_tanh_f32)
  return __builtin_amdgcn_tanh_f32(x);
#else
  return tanhf(x);
#endif
}

// ---------------------------------------------------------------------------
// Pack kernel: build WMMA A-operand tables in workspace.
//
// tabA1[t*32+lane]  (v2f):  A1 tile for GEMM1 (16x4, M=hidden, K=[w1x,w1y,b1,0])
//   lanes 0-15 : (w1x_j, w1y_j)  K=0,1      j = 16t + (lane&15)
//   lanes16-31 : (b1_j , 0   )   K=2,3
//
// tabA2[(t*4+p)*32+lane] (v2f): A2 tile for GEMM2 (16x4, M=output row, K=hidden
//   indices {r, r+1, r+8, r+9} with r=2p, matching the D->B half-wave layout)
//   lane 0  : (pu_{j0}, pu_{j0+1})   lane 16 : (pu_{j0+8}, pu_{j0+9})
//   lane 1  : (pv_{j0}, pv_{j0+1})   lane 17 : (pv_{j0+8}, pv_{j0+9})
//   others  : (0,0)                  j0 = 16t + 2p
// ---------------------------------------------------------------------------
__global__ void kPackWeights(const float* __restrict__ W1, const float* __restrict__ b1,
                             const float* __restrict__ W2,
                             v2f* __restrict__ tabA1, v2f* __restrict__ tabA2) {
  int idx = blockIdx.x * blockDim.x + threadIdx.x;
  const int totalA1 = NTILE * 32;
  const int totalA2 = NTILE * 4 * 32;
  if (idx < totalA1) {
    int t = idx >> 5, lane = idx & 31;
    int j = 16 * t + (lane & 15);
    v2f v = {0.0f, 0.0f};
    if (j < HID) {
      if (lane < 16) { v.x = W1[j];  v.y = W1[HID + j]; }  // (w1x_j, w1y_j)
      else           { v.x = b1[j];  v.y = 0.0f;        }  // (b1_j, 0)
    }
    tabA1[idx] = v;
  } else if (idx < totalA1 + totalA2) {
    int k = idx - totalA1;
    int tp = k >> 5, lane = k & 31;
    int t = tp >> 2, p = tp & 3;
    int half = (lane < 16) ? 0 : 8;
    int m = lane & 15;
    int j0 = 16 * t + 2 * p + half;
    v2f v = {0.0f, 0.0f};
    if (m == 0 || m == 1) {
      float vals[2];
      for (int e = 0; e < 2; ++e) {
        int j = j0 + e;
        float val = 0.0f;
        if (j < HID) {
          float wx = W1[j], wy = W1[HID + j];
          float g = (m == 0) ? (wx * wy - wy * wy)    // pu: Hxy - Hyy
                             : (wx * wy - wx * wx);   // pv: Hxy - Hxx
          val = -2.0f * W2[j] * g;
        }
        vals[e] = val;
      }
      v.x = vals[0]; v.y = vals[1];
    }
    tabA2[k] = v;
  }
}

// ---------------------------------------------------------------------------
// Main kernel: one 16-point tile per wave iteration, grid-stride over tiles.
// ---------------------------------------------------------------------------
__global__ void __launch_bounds__(256)
kDivFreeHess(const v2f* __restrict__ xy, const v2f* __restrict__ tabA1,
             const v2f* __restrict__ tabA2, v2f* __restrict__ out,
             int ntiles, int npts) {
  const int lane   = threadIdx.x & 31;
  const int wave   = (blockIdx.x * blockDim.x + threadIdx.x) >> 5;
  const int nwaves = (gridDim.x * blockDim.x) >> 5;

  // Hoist all weight operand tiles into registers (broadcast; L2/WGP$-resident)
  v2f A1[NTILE];
  v2f A2[NTILE][4];
#pragma unroll
  for (int t = 0; t < NTILE; ++t) {
    A1[t] = tabA1[t * 32 + lane];
#pragma unroll
    for (int p = 0; p < 4; ++p)
      A2[t][p] = tabA2[(t * 4 + p) * 32 + lane];
  }

  const v8f zero8 = (v8f)(0.0f);

  for (int tile = wave; tile < ntiles; tile += nwaves) {
    // B1 (4x16): rows [x; y; 1; 0], columns = 16 points of this tile.
    int pidx = tile * 16 + (lane & 15);
    pidx = pidx < npts ? pidx : npts - 1;
    v2f q = xy[pidx];
    v2f B1;
    B1.x = (lane < 16) ? q.x : 1.0f;   // K=0 row x   | K=2 row ones
    B1.y = (lane < 16) ? q.y : 0.0f;   // K=1 row y   | K=3 row zeros

    v8f C0 = zero8, C1 = zero8;        // two chains: row0 = u, row1 = v
#pragma unroll
    for (int t = 0; t < NTILE; ++t) {
      // GEMM1: a-tile^T (16 hidden x 16 points) = A1 x B1, exact f32
      v8f a = __builtin_amdgcn_wmma_f32_16x16x4_f32(
          false, A1[t], false, B1, (short)0, zero8, false, false);

      // s = tanh(a) * (1 - tanh(a)^2), elementwise on accumulator layout
      v8f s;
#pragma unroll
      for (int i = 0; i < 8; ++i) {
        float h = fast_tanh(a[i]);
        s[i] = h * __builtin_fmaf(-h, h, 1.0f);
      }

      // GEMM2: accumulate (u,v) — VGPR pair (s_{2p}, s_{2p+1}) is a valid
      // B-operand covering hidden K = {2p, 2p+1, 2p+8, 2p+9}; A2 matches.
#pragma unroll
      for (int p = 0; p < 4; ++p) {
        v2f B2; B2.x = s[2 * p]; B2.y = s[2 * p + 1];
        if (t & 1)
          C1 = __builtin_amdgcn_wmma_f32_16x16x4_f32(
              false, A2[t][p], false, B2, (short)0, C1, false, false);
        else
          C0 = __builtin_amdgcn_wmma_f32_16x16x4_f32(
              false, A2[t][p], false, B2, (short)0, C0, false, false);
      }
    }

    // D rows: row0 (VGPR0) = u, row1 (VGPR1) = v, for point = lane (0..15)
    int pt = tile * 16 + lane;
    if (lane < 16 && pt < npts) {
      v2f r; r.x = C0[0] + C1[0]; r.y = C0[1] + C1[1];
      out[pt] = r;
    }
  }
}

extern "C" void kernel_launch(void* const* d_in, const int* in_sizes, int n_in,
                              void* d_out, int out_size, void* d_ws, size_t ws_size,
                              hipStream_t stream) {
  // setup_inputs order: f, xy, W1, b1, W2, b2
  const v2f*   xy = (const v2f*)d_in[1];
  const float* W1 = (const float*)d_in[2];
  const float* b1 = (const float*)d_in[3];
  const float* W2 = (const float*)d_in[4];

  int npts   = in_sizes[1] / 2;
  int ntiles = (npts + 15) / 16;

  v2f* tabA1 = (v2f*)d_ws;                 // 7*32   v2f = 1792 B
  v2f* tabA2 = tabA1 + NTILE * 32;         // 28*32  v2f = 7168 B

  int packTotal = NTILE * 32 + NTILE * 4 * 32;
  kPackWeights<<<(packTotal + 255) / 256, 256, 0, stream>>>(W1, b1, W2, tabA1, tabA2);

  int blocks = 1024;                       // 8192 waves, ~4 tiles each
  kDivFreeHess<<<blocks, 256, 0, stream>>>(xy, tabA1, tabA2, (v2f*)d_out,
                                           ntiles, npts);
}